// CirModel_52956946760343
// MI455X (gfx1250) — compile-verified
//
#include <hip/hip_runtime.h>
#include <math.h>

typedef float v2f __attribute__((ext_vector_type(2)));
typedef float v8f __attribute__((ext_vector_type(8)));

#define NQ      14
#define DIM     16384
#define THREADS 512
#define W_MUL   0.63245553203367587f   // sqrt(2)/sqrt(5)
#define USE_ASYNC_LDS 1

// ---------------------------------------------------------------------------
// Ring-of-CNOT = GF(2)-linear permutation of basis indices.
// gather:  new[y] = old[Mg y], Mg = C(0,1)*C(1,2)*...*C(13,0)  (rightmost first)
// Masks computed at compile time, expressed over *state-bit* positions
// (qubit q <-> state bit 13-q, since axis 1 of the reshape is the MSB).
// ---------------------------------------------------------------------------
struct PermMasks { unsigned m[NQ]; };
static constexpr PermMasks make_perm() {
    PermMasks pm{};
    unsigned P[NQ] = {};
    for (int i = 0; i < NQ; ++i) P[i] = 1u << i;
    for (int k = NQ - 1; k >= 0; --k) { int c = k, t = (k + 1) % NQ; P[t] ^= P[c]; }
    for (int b = 0; b < NQ; ++b) {
        unsigned src = P[NQ - 1 - b], s = 0;
        for (int q = 0; q < NQ; ++q) if ((src >> q) & 1u) s |= 1u << (NQ - 1 - q);
        pm.m[b] = s;
    }
    return pm;
}

// ---------------------------------------------------------------------------
// Gate-matrix precompute: 20 blocks = (5 layers x 4 groups), 256 thr = 16x16.
// Fuses the 3 rotations per qubit into one 2x2, then tensor-products 4 qubits
// into a 16x16 complex matrix.  ws layout: Gr[20][256] then Gi[20][256].
// ---------------------------------------------------------------------------
struct C2 { float re, im; };
__device__ inline C2 cmul(C2 a, C2 b) { return C2{a.re*b.re - a.im*b.im, a.re*b.im + a.im*b.re}; }
__device__ inline C2 cadd(C2 a, C2 b) { return C2{a.re + b.re, a.im + b.im}; }
struct M2 { C2 e[2][2]; };
__device__ inline M2 mmul2(const M2& x, const M2& y) {
    M2 r;
    #pragma unroll
    for (int i = 0; i < 2; ++i)
        #pragma unroll
        for (int j = 0; j < 2; ++j)
            r.e[i][j] = cadd(cmul(x.e[i][0], y.e[0][j]), cmul(x.e[i][1], y.e[1][j]));
    return r;
}
__device__ inline M2 mk_gate(int type, float th) {
    float h = 0.5f * th;
    float c = cosf(h), s = sinf(h);
    M2 m;
    if (type == 0) {            // RX
        m.e[0][0] = C2{c, 0.f};  m.e[0][1] = C2{0.f, -s};
        m.e[1][0] = C2{0.f, -s}; m.e[1][1] = C2{c, 0.f};
    } else if (type == 1) {     // RY
        m.e[0][0] = C2{c, 0.f};  m.e[0][1] = C2{-s, 0.f};
        m.e[1][0] = C2{s, 0.f};  m.e[1][1] = C2{c, 0.f};
    } else {                    // RZ
        m.e[0][0] = C2{c, -s};   m.e[0][1] = C2{0.f, 0.f};
        m.e[1][0] = C2{0.f, 0.f}; m.e[1][1] = C2{c, s};
    }
    return m;
}

__global__ __launch_bounds__(256) void gate_precompute(const float* __restrict__ w,
                                                       float* __restrict__ g) {
    const int li  = blockIdx.x >> 2;   // layer 0..4
    const int grp = blockIdx.x & 3;    // group 0..3
    const int t = threadIdx.x;
    const int r = t >> 4, c = t & 15;
    const int gt[5][3] = {{0,1,2},{1,2,1},{2,1,0},{0,2,0},{1,2,1}}; // 0=RX 1=RY 2=RZ
    const int qa = (grp < 3) ? grp * 4 : 10;   // group 3: qubits 10..13, 10/11 = identity
    C2 prod = C2{1.f, 0.f};
    #pragma unroll
    for (int j = 0; j < 4; ++j) {
        const int q = qa + j;
        M2 m;
        if (grp == 3 && j < 2) {
            m.e[0][0] = C2{1.f,0.f}; m.e[0][1] = C2{0.f,0.f};
            m.e[1][0] = C2{0.f,0.f}; m.e[1][1] = C2{1.f,0.f};
        } else {
            M2 g0 = mk_gate(gt[li][0], w[42*li + 3*q + 0] * W_MUL);
            M2 g1 = mk_gate(gt[li][1], w[42*li + 3*q + 1] * W_MUL);
            M2 g2 = mk_gate(gt[li][2], w[42*li + 3*q + 2] * W_MUL);
            m = mmul2(g2, mmul2(g1, g0));   // applied g0 first
        }
        const int rb = (r >> (3 - j)) & 1, cb = (c >> (3 - j)) & 1;
        prod = cmul(prod, m.e[rb][cb]);
    }
    g[(blockIdx.x << 8) + t]        = prod.re;
    g[5120 + (blockIdx.x << 8) + t] = prod.im;
}

// insert 4-bit group value gv at bit position p of the 14-bit index
__device__ inline int ins_bits(int gv, int col, int p) {
    const int low  = col & ((1 << p) - 1);
    const int high = col >> p;
    return (high << (p + 4)) | (gv << p) | low;
}

// ---------------------------------------------------------------------------
// Main kernel: one workgroup (16 waves) per batch element.  State lives in
// LDS; each layer = 4 x (16x16 complex) group-applies via V_WMMA_F32_16X16X4_F32.
// ---------------------------------------------------------------------------
__global__ __launch_bounds__(THREADS) void qsim(const float* __restrict__ xr,
                                                const float* __restrict__ xi,
                                                const float* __restrict__ gm,
                                                float* __restrict__ out) {
    __shared__ __align__(16) float sRe[DIM];
    __shared__ __align__(16) float sIm[DIM];
    __shared__ float sred[16];

    const int tid  = threadIdx.x;
    const int lane = tid & 31;
    const int wave = tid >> 5;
    const int half = lane >> 4;      // 0: lanes 0-15, 1: lanes 16-31
    const int lrow = lane & 15;
    const long base = (long)blockIdx.x * DIM;

    // ---- load state into LDS ----
#if USE_ASYNC_LDS
    {
        // CDNA5 async copy: HBM -> LDS without VGPR staging (ASYNCcnt-tracked)
        const unsigned lre = (unsigned)(uintptr_t)(void*)sRe;
        const unsigned lim = (unsigned)(uintptr_t)(void*)sIm;
        #pragma unroll
        for (int j = 0; j < 8; ++j) {
            const int e = tid + j * THREADS;              // float4 index
            const unsigned la_r = lre + (unsigned)e * 16u;
            const unsigned la_i = lim + (unsigned)e * 16u;
            const void* ga_r = (const void*)(xr + base + (long)e * 4);
            const void* ga_i = (const void*)(xi + base + (long)e * 4);
            asm volatile("global_load_async_to_lds_b128 %0, %1, off"
                         :: "v"(la_r), "v"(ga_r) : "memory");
            asm volatile("global_load_async_to_lds_b128 %0, %1, off"
                         :: "v"(la_i), "v"(ga_i) : "memory");
        }
        asm volatile("s_wait_asynccnt 0x0" ::: "memory");
    }
#else
    {
        const float4* xr4 = (const float4*)(xr + base);
        const float4* xi4 = (const float4*)(xi + base);
        float4* sR4 = (float4*)sRe;
        float4* sI4 = (float4*)sIm;
        #pragma unroll
        for (int j = 0; j < 8; ++j) {
            sR4[tid + j * THREADS] = xr4[tid + j * THREADS];
            sI4[tid + j * THREADS] = xi4[tid + j * THREADS];
        }
    }
#endif
    __syncthreads();

    for (int li = 0; li < 5; ++li) {
        #pragma unroll 1
        for (int g = 0; g < 4; ++g) {
            const int p = (g < 3) ? (10 - 4 * g) : 0;   // group bit position
            const float* Gr = gm + ((li * 4 + g) << 8);
            const float* Gi = Gr + 5120;
            // A fragments (16x4 f32 layout): lane = row M, K = kt*4 + 2*half + elem
            v2f ar[4], ai[4], ain[4];
            const int ab = lrow * 16 + 2 * half;
            #pragma unroll
            for (int kt = 0; kt < 4; ++kt) {
                ar[kt].x = Gr[ab + kt * 4];     ar[kt].y = Gr[ab + kt * 4 + 1];
                ai[kt].x = Gi[ab + kt * 4];     ai[kt].y = Gi[ab + kt * 4 + 1];
                ain[kt]  = -ai[kt];             // for Dr -= Ai*Bi (f32 WMMA has no A-neg)
            }
            #pragma unroll 1
            for (int j = 0; j < 4; ++j) {       // 4 column tiles per wave (64 total)
                const int col = (wave * 4 + j) * 16 + lrow;
                v2f br[4], bi[4];
                #pragma unroll
                for (int kt = 0; kt < 4; ++kt) {
                    const int gv0 = kt * 4 + 2 * half;
                    const int f0 = ins_bits(gv0,     col, p);
                    const int f1 = ins_bits(gv0 + 1, col, p);
                    br[kt].x = sRe[f0]; br[kt].y = sRe[f1];
                    bi[kt].x = sIm[f0]; bi[kt].y = sIm[f1];
                }
                v8f dr = {0,0,0,0,0,0,0,0};
                v8f di = {0,0,0,0,0,0,0,0};
                #pragma unroll
                for (int kt = 0; kt < 4; ++kt) {
                    dr = __builtin_amdgcn_wmma_f32_16x16x4_f32(false, ar[kt],  false, br[kt], (short)0, dr, false, false);
                    dr = __builtin_amdgcn_wmma_f32_16x16x4_f32(false, ain[kt], false, bi[kt], (short)0, dr, false, false);
                    di = __builtin_amdgcn_wmma_f32_16x16x4_f32(false, ar[kt],  false, bi[kt], (short)0, di, false, false);
                    di = __builtin_amdgcn_wmma_f32_16x16x4_f32(false, ai[kt],  false, br[kt], (short)0, di, false, false);
                }
                // D layout: element i, half h -> M = i + 8h, N = col
                #pragma unroll
                for (int i = 0; i < 8; ++i) {
                    const int f = ins_bits(i + 8 * half, col, p);
                    sRe[f] = dr[i];
                    sIm[f] = di[i];
                }
            }
            __syncthreads();
        }
        if (li < 4) {  // ring of CNOT = one permutation pass
            constexpr PermMasks pm = make_perm();
            float tr[32], ti[32];
            #pragma unroll
            for (int j = 0; j < 32; ++j) {
                const int y = tid + j * THREADS;
                int x = 0;
                #pragma unroll
                for (int b = 0; b < NQ; ++b)
                    x |= (__popc(pm.m[b] & (unsigned)y) & 1) << b;
                tr[j] = sRe[x]; ti[j] = sIm[x];
            }
            __syncthreads();
            #pragma unroll
            for (int j = 0; j < 32; ++j) {
                const int y = tid + j * THREADS;
                sRe[y] = tr[j]; sIm[y] = ti[j];
            }
            __syncthreads();
        }
    }

    // ---- <Z_0> : sign by state bit 13 ----
    float acc = 0.f;
    #pragma unroll
    for (int j = 0; j < 32; ++j) {
        const int idx = tid + j * THREADS;
        const float re = sRe[idx], im = sIm[idx];
        const float v = re * re + im * im;
        acc += (idx & 0x2000) ? -v : v;
    }
    #pragma unroll
    for (int o = 16; o > 0; o >>= 1) acc += __shfl_down(acc, o, 32);
    if (lane == 0) sred[wave] = acc;
    __syncthreads();
    if (tid == 0) {
        float s = 0.f;
        #pragma unroll
        for (int wv = 0; wv < 16; ++wv) s += sred[wv];
        out[blockIdx.x] = s;
    }
}

// ---------------------------------------------------------------------------
extern "C" void kernel_launch(void* const* d_in, const int* in_sizes, int n_in,
                              void* d_out, int out_size, void* d_ws, size_t ws_size,
                              hipStream_t stream) {
    const float* xr = (const float*)d_in[0];
    const float* xi = (const float*)d_in[1];
    const float* w  = (const float*)d_in[2];
    float* out = (float*)d_out;
    float* gm  = (float*)d_ws;                 // 10240 floats = 40 KB
    const int B = in_sizes[0] / DIM;

    gate_precompute<<<20, 256, 0, stream>>>(w, gm);
    qsim<<<B, THREADS, 0, stream>>>(xr, xi, gm, out);
    (void)n_in; (void)out_size; (void)ws_size;
}